// PointTransientRenderer_44014824849467
// MI455X (gfx1250) — compile-verified
//
#include <hip/hip_runtime.h>

typedef float v2f __attribute__((ext_vector_type(2)));
typedef float v8f __attribute__((ext_vector_type(8)));

#define HNUM 64
#define WNUM 64
#define TNUM 2048
#define BNUM 64
#define PNUM 512
#define VDIM 128
#define KNUM 512   // TNUM/4, conv kernel length

// voxel value of the masked/clipped albedo volume (clip BEFORE interpolation,
// matching the reference)
__device__ __forceinline__ float voxel_val(const float* __restrict__ pv,
                                           const float* __restrict__ pa,
                                           int idx) {
  float v = pv[idx];
  if (v > 0.0f) {
    float s = pa[idx] + v;
    return fminf(fmaxf(s, 0.0f), 1.0f);
  }
  return 0.0f;
}

// ---------------------------------------------------------------------------
// Kernel A: per-(b,p) radiometric value and path length.
// laser_pos == camera_pos == wall grid point, so sv==lv (sd=ld, svn=lvn).
// ---------------------------------------------------------------------------
__global__ void k_value_dist(const float* __restrict__ surf_pos,
                             const float* __restrict__ surf_nrm,
                             const int*   __restrict__ xys,
                             const float* __restrict__ pv,
                             const float* __restrict__ pa,
                             const float* __restrict__ scale_p,
                             float* __restrict__ value,
                             float* __restrict__ dist) {
  int idx = blockIdx.x * blockDim.x + threadIdx.x;  // 0 .. B*P-1
  int b = idx >> 9;            // / 512
  const float* sp = surf_pos + (size_t)idx * 3;
  float px = sp[0], py = sp[1], pz = sp[2];
  const float* sn = surf_nrm + (size_t)idx * 3;
  float nx = sn[0], ny = sn[1], nz = sn[2];

  int yi = xys[b * 2 + 0];
  int xi = xys[b * 2 + 1];
  float wxp = -1.0f + 2.0f * (float)xi / 63.0f;   // wall x
  float wyp = -1.0f + 2.0f * (float)yi / 63.0f;   // wall y
  // d1: wall -> laser origin (0,0,-1); d4: wall -> camera origin (0.05,0,-1)
  float d1 = sqrtf(wxp * wxp + wyp * wyp + 1.0f);
  float cx = wxp - 0.05f;
  float d4 = sqrtf(cx * cx + wyp * wyp + 1.0f);

  // trilinear sample of vol[z,y,x] = albedo_vol[y,x,z], grid = (x, -y, z-1)
  float fx = (px + 1.0f) * 0.5f * 63.0f;
  float fy = (-py + 1.0f) * 0.5f * 63.0f;
  float fz = ((pz - 1.0f) + 1.0f) * 0.5f * 127.0f;
  float x0f = floorf(fx), y0f = floorf(fy), z0f = floorf(fz);
  float wx = fx - x0f, wy = fy - y0f, wz = fz - z0f;
  int x0 = (int)x0f, y0 = (int)y0f, z0 = (int)z0f;

  float alb = 0.0f;
  #pragma unroll
  for (int dz = 0; dz < 2; ++dz)
    #pragma unroll
    for (int dy = 0; dy < 2; ++dy)
      #pragma unroll
      for (int dx = 0; dx < 2; ++dx) {
        int ix = x0 + dx, iy = y0 + dy, iz = z0 + dz;
        bool valid = (ix >= 0) && (ix < WNUM) && (iy >= 0) && (iy < HNUM) &&
                     (iz >= 0) && (iz < VDIM);
        int ccx = min(max(ix, 0), WNUM - 1);
        int ccy = min(max(iy, 0), HNUM - 1);
        int ccz = min(max(iz, 0), VDIM - 1);
        float v = voxel_val(pv, pa, (ccy * WNUM + ccx) * VDIM + ccz);
        float w = (dx ? wx : 1.0f - wx) * (dy ? wy : 1.0f - wy) *
                  (dz ? wz : 1.0f - wz);
        alb += (valid ? v : 0.0f) * w;
      }

  float lx = wxp - px, ly = wyp - py, lz = -pz;
  float ld = sqrtf(lx * lx + ly * ly + lz * lz);
  float inv = 1.0f / fmaxf(ld, 1e-12f);
  float lnx = lx * inv, lny = ly * inv, lnz = lz * inv;
  float dl = lnx * nx + lny * ny + lnz * nz;

  float scale = scale_p[0];
  // value = scale*alb/(ld^2*sd^2) * (-lnz)(-snz) * dot_l * dot_s ; sv==lv
  float val = scale * alb / (ld * ld * ld * ld) * (lnz * lnz) * (dl * dl);
  value[idx] = val;
  dist[idx] = d1 + d4 + 2.0f * ld;
}

// ---------------------------------------------------------------------------
// Kernel B: tdata[b,t] = sum_p value * exp(-(t*BIN - dist)^2 / (2*sig2))
// Transcendental-bound; stage value/dist rows in LDS.
// ---------------------------------------------------------------------------
__global__ void k_tdata(const float* __restrict__ value,
                        const float* __restrict__ dist,
                        float* __restrict__ tdata) {
  __shared__ float s_val[PNUM];
  __shared__ float s_dst[PNUM];
  int b = blockIdx.y;
  int tid = threadIdx.x;                // 256 threads
  s_val[tid]       = value[b * PNUM + tid];
  s_val[tid + 256] = value[b * PNUM + tid + 256];
  s_dst[tid]       = dist[b * PNUM + tid];
  s_dst[tid + 256] = dist[b * PNUM + tid + 256];
  __syncthreads();

  int t = blockIdx.x * 256 + tid;
  float tt = (float)t * 0.01f;
  const float INV2SIG2 = 27725.887222397812f;  // 1/(2*sig2), sig2=1e-4/(8 ln2)
  float sum = 0.0f;
  #pragma unroll 4
  for (int p = 0; p < PNUM; ++p) {
    float d = tt - s_dst[p];
    sum += s_val[p] * expf(-d * d * INV2SIG2);
  }
  tdata[b * TNUM + t] = sum;
}

// ---------------------------------------------------------------------------
// Kernel C: ir[j] = same-mode conv of gauss (sigma) with one-sided exp (lamda)
// ir[j] = full[j+255] = sum_i gauss[i] * resp[j+255-i]
// ---------------------------------------------------------------------------
__global__ void k_ir(const float* __restrict__ sigma_p,
                     const float* __restrict__ lamda_p,
                     float* __restrict__ ir) {
  int j = threadIdx.x;  // 0..511
  float sinv = 1.0f / sigma_p[0];
  float lam = lamda_p[0];
  const float C = 0.3989422804014327f;  // 1/sqrt(2*pi)
  float sum = 0.0f;
  for (int i = 0; i < KNUM; ++i) {
    int k = j + 255 - i;
    if (k < 0 || k >= KNUM) continue;
    float at = (float)(i - 256);
    float g = sinv * C * expf(-at * at * sinv * sinv * 0.5f);
    float r = (k >= 256) ? lam * expf(-lam * (float)(k - 256)) : 0.0f;
    sum += g * r;
  }
  ir[j] = sum;
}

// ---------------------------------------------------------------------------
// Kernel D: banded-Toeplitz GEMM via V_WMMA_F32_16X16X4_F32 (wave32).
//   Out[b,t] = dc + sum_s X[b,s] * ir[t+255-s],  s in [t-256, t+255]
// One wave per 16-wide t-tile; M=64 batches = 4 WMMA M-tiles; K band = 33
// chunks of 16 (4 WMMAs each).  f32 WMMA == FMA-class numerics.
// ---------------------------------------------------------------------------
__global__ void __launch_bounds__(32)
k_conv_wmma(const float* __restrict__ tdata,
            const float* __restrict__ ir,
            const float* __restrict__ dc_p,
            float* __restrict__ out) {
  __shared__ float s_ir[KNUM];
  int lane = threadIdx.x;               // 0..31
  #pragma unroll
  for (int i = 0; i < KNUM / 32; ++i) s_ir[lane + i * 32] = ir[lane + i * 32];
  __syncthreads();

  int t0 = blockIdx.x * 16;
  int half = lane >> 4;                 // 0: lanes 0-15, 1: lanes 16-31
  int r = lane & 15;
  float dc = dc_p[0];

  v8f acc[4];
  #pragma unroll
  for (int m = 0; m < 4; ++m) acc[m] = (v8f){};

  for (int c = 0; c < 33; ++c) {        // K band: s in [t0-256, t0+271]
    int s_chunk = t0 - 256 + c * 16;
    #pragma unroll
    for (int kk = 0; kk < 4; ++kk) {
      int sb = s_chunk + kk * 4;
      // A layout: lanes 0-15 K=0,1 ; lanes 16-31 K=2,3  (row = r)
      // B layout: lanes 0-15 K=0,1 ; lanes 16-31 K=2,3  (col = r)
      int s0 = sb + half * 2;
      int s1 = s0 + 1;

      int i0 = t0 + r + 255 - s0;       // H[s,t] = ir[t+255-s]
      int i1 = t0 + r + 255 - s1;
      v2f bf;
      bf.x = (i0 >= 0 && i0 < KNUM) ? s_ir[i0] : 0.0f;
      bf.y = (i1 >= 0 && i1 < KNUM) ? s_ir[i1] : 0.0f;

      bool v0 = (s0 >= 0) && (s0 < TNUM);
      bool v1 = (s1 >= 0) && (s1 < TNUM);
      #pragma unroll
      for (int m = 0; m < 4; ++m) {
        int row = m * 16 + r;
        v2f af;
        af.x = v0 ? tdata[row * TNUM + s0] : 0.0f;
        af.y = v1 ? tdata[row * TNUM + s1] : 0.0f;
        acc[m] = __builtin_amdgcn_wmma_f32_16x16x4_f32(
            false, af, false, bf, (short)0, acc[m], false, false);
      }
    }
  }

  // D layout: VGPR j -> row M = j + half*8, col N = r
  #pragma unroll
  for (int m = 0; m < 4; ++m)
    #pragma unroll
    for (int j = 0; j < 8; ++j) {
      int row = m * 16 + half * 8 + j;
      out[row * TNUM + t0 + r] = acc[m][j] + dc;
    }
}

// ---------------------------------------------------------------------------
extern "C" void kernel_launch(void* const* d_in, const int* in_sizes, int n_in,
                              void* d_out, int out_size, void* d_ws,
                              size_t ws_size, hipStream_t stream) {
  const float* surf_pos = (const float*)d_in[0];
  const float* surf_nrm = (const float*)d_in[1];
  const int*   xys      = (const int*)d_in[2];
  const float* pv       = (const float*)d_in[3];
  const float* pa       = (const float*)d_in[4];
  const float* scale_p  = (const float*)d_in[5];
  const float* sigma_p  = (const float*)d_in[6];
  const float* lamda_p  = (const float*)d_in[7];
  const float* dc_p     = (const float*)d_in[8];

  float* out       = (float*)d_out;
  float* tdata_out = out;                 // first B*TNUM floats
  float* tdata     = out + BNUM * TNUM;   // second B*TNUM floats

  float* ws    = (float*)d_ws;
  float* value = ws;                      // B*P
  float* dstv  = ws + BNUM * PNUM;        // B*P
  float* ir    = ws + 2 * BNUM * PNUM;    // KNUM

  k_value_dist<<<(BNUM * PNUM) / 256, 256, 0, stream>>>(
      surf_pos, surf_nrm, xys, pv, pa, scale_p, value, dstv);

  dim3 g2(TNUM / 256, BNUM);
  k_tdata<<<g2, 256, 0, stream>>>(value, dstv, tdata);

  k_ir<<<1, KNUM, 0, stream>>>(sigma_p, lamda_p, ir);

  k_conv_wmma<<<TNUM / 16, 32, 0, stream>>>(tdata, ir, dc_p, tdata_out);
}